// CheckNodeTrellis_16303695856317
// MI455X (gfx1250) — compile-verified
//
#include <hip/hip_runtime.h>
#include <hip/hip_bf16.h>
#include <math.h>
#include <stdint.h>

typedef __attribute__((ext_vector_type(2))) float v2f;
typedef __attribute__((ext_vector_type(8))) float v8f;

// out[b, i, s, k2] = logsumexp_{j,k} ( e1[b, (i+j)%2, s, k] + e2[b, j, k, k2] )
// Exp-space f32 WMMA matmuls + per-element log recombine.
// One workgroup (8 x wave32) per batch b. Global->LDS fill uses the CDNA5
// async direct-to-LDS path (ASYNCcnt) instead of staging through VGPRs.
__global__ __launch_bounds__(256) void trellis_lse_wmma(const float* __restrict__ e1,
                                                        const float* __restrict__ e2,
                                                        float* __restrict__ out) {
    // exp(e1 - rowmax):  Ae[p][s][k]
    // exp(e2 - colmax):  Be[j][k][k2]
    __shared__ float Ae[2][64][64];
    __shared__ float Be[2][64][64];
    __shared__ float m1s[2][64];   // rowmax of e1 per parity p, row s
    __shared__ float m2s[2][64];   // colmax of e2 per j, column k2

    const int b   = blockIdx.x;      // batch index b1*16 + b2, 256 total
    const int tid = threadIdx.x;     // 0..255

    const float* e1b = e1 + (size_t)b * (2 * 64 * 64);
    const float* e2b = e2 + (size_t)b * (2 * 64 * 64);
    float*       ob  = out + (size_t)b * (2 * 64 * 64);

    // ---- Phase 1a: async global -> LDS fill (no VGPR staging) ----
    // Each lane moves 16B per instruction; 8 iters x 2 tensors covers
    // 2 x 32 KiB. Tracked by ASYNCcnt; single wait before the barrier.
    {
        // Generic shared-memory addresses carry the LDS byte offset in the
        // low 32 bits (aperture in the high bits) -> truncate to get dsaddr.
        const uint32_t la = (uint32_t)(uintptr_t)&Ae[0][0][0];
        const uint32_t lb = (uint32_t)(uintptr_t)&Be[0][0][0];
#pragma unroll
        for (int r = 0; r < 8; ++r) {
            const int fidx = (tid + r * 256) * 4;     // float index, 16B aligned
            const uint32_t da = la + (uint32_t)fidx * 4u;
            const uint32_t db = lb + (uint32_t)fidx * 4u;
            const float* ga = e1b + fidx;
            const float* gb = e2b + fidx;
            asm volatile("global_load_async_to_lds_b128 %0, %1, off"
                         :: "v"(da), "v"(ga) : "memory");
            asm volatile("global_load_async_to_lds_b128 %0, %1, off"
                         :: "v"(db), "v"(gb) : "memory");
        }
        asm volatile("s_wait_asynccnt 0x0" ::: "memory");
    }
    __syncthreads();

    // ---- Phase 1b: row maxes of e1 (tid<128) / column maxes of e2 (tid>=128) ----
    if (tid < 128) {
        const int p = tid >> 6, s = tid & 63;
        float m = -INFINITY;
#pragma unroll 8
        for (int k = 0; k < 64; ++k) m = fmaxf(m, Ae[p][s][k]);
        m1s[p][s] = m;
    } else {
        const int t = tid - 128;
        const int j = t >> 6, n = t & 63;
        float m = -INFINITY;
#pragma unroll 8
        for (int k = 0; k < 64; ++k) m = fmaxf(m, Be[j][k][n]);
        m2s[j][n] = m;
    }
    __syncthreads();

    // ---- Phase 1c: exponentiate in place (32 elements per thread per tensor) ----
    {
        float* fA = &Ae[0][0][0];
        float* fB = &Be[0][0][0];
#pragma unroll
        for (int r = 0; r < 32; ++r) {
            const int e = tid + r * 256;          // 0..8191
            const int hi = e >> 12;               // parity p  (Ae) / j (Be)
            const int mid = (e >> 6) & 63;        // row s     (Ae) / k (Be)
            const int lo = e & 63;                // k         (Ae) / k2 (Be)
            fA[e] = __expf(fA[e] - m1s[hi][mid]);
            fB[e] = __expf(fB[e] - m2s[hi][lo]);
        }
    }
    __syncthreads();

    // ---- Phase 2: WMMA f32 16x16x4 accumulation, two chains per output tile ----
    const int lane  = tid & 31;
    const int wave  = tid >> 5;     // 0..7
    const int i     = wave >> 2;    // output parity slice
    const int st    = wave & 3;     // s-tile (rows st*16 .. st*16+15)
    const int p0    = i;            // parity used by chain j=0
    const int p1    = 1 - i;        // parity used by chain j=1

    // f32 16x16x4 fragment lane mapping (ISA 7.12.2):
    //   A: lane<16 -> M=lane,    K = {k0, k0+1};  lane>=16 -> M=lane-16, K = {k0+2, k0+3}
    //   B: same K split across lane halves; N = lane & 15
    const int mrow  = lane & 15;
    const int khalf = lane >> 4;

    for (int k2t = 0; k2t < 4; ++k2t) {
        v8f c0 = {};   // chain j=0 accumulator
        v8f c1 = {};   // chain j=1 accumulator
        const int n = k2t * 16 + mrow;
#pragma unroll
        for (int kk = 0; kk < 16; ++kk) {
            const int kb = kk * 4 + 2 * khalf;    // this lane's K pair base
            v2f a0, a1, b0, b1;
            a0.x = Ae[p0][st * 16 + mrow][kb];
            a0.y = Ae[p0][st * 16 + mrow][kb + 1];
            a1.x = Ae[p1][st * 16 + mrow][kb];
            a1.y = Ae[p1][st * 16 + mrow][kb + 1];
            b0.x = Be[0][kb][n];
            b0.y = Be[0][kb + 1][n];
            b1.x = Be[1][kb][n];
            b1.y = Be[1][kb + 1][n];
            c0 = __builtin_amdgcn_wmma_f32_16x16x4_f32(false, a0, false, b0,
                                                       (short)0, c0, false, false);
            c1 = __builtin_amdgcn_wmma_f32_16x16x4_f32(false, a1, false, b1,
                                                       (short)0, c1, false, false);
        }

        // ---- log-recombine:  out = M + log( e^{t0-M} c0 + e^{t1-M} c1 ) ----
#pragma unroll
        for (int r = 0; r < 8; ++r) {
            const int srow = st * 16 + r + 8 * khalf;   // C/D layout: M = r (+8 upper half)
            const int col  = k2t * 16 + mrow;           // N = lane & 15
            const float t0 = m1s[p0][srow] + m2s[0][col];
            const float t1 = m1s[p1][srow] + m2s[1][col];
            const float mm = fmaxf(t0, t1);
            const float v  = mm + __logf(__expf(t0 - mm) * c0[r] +
                                         __expf(t1 - mm) * c1[r]);
            ob[((size_t)i * 64 + srow) * 64 + col] = v;
        }
    }
}

extern "C" void kernel_launch(void* const* d_in, const int* in_sizes, int n_in,
                              void* d_out, int out_size, void* d_ws, size_t ws_size,
                              hipStream_t stream) {
    const float* e1 = (const float*)d_in[0];
    const float* e2 = (const float*)d_in[1];
    float* out = (float*)d_out;
    trellis_lse_wmma<<<dim3(256), dim3(256), 0, stream>>>(e1, e2, out);
}